// ImageSatistics_68058051772648
// MI455X (gfx1250) — compile-verified
//
#include <hip/hip_runtime.h>
#include <math.h>

// ---------------- problem constants ----------------
#define C_     12
#define B_     64
#define HW_    65536            // H*W
#define SLICES (B_ * C_)        // 768 contiguous (b,c) slices of 65536 floats
#define BINS_  1000
#define NFINE  4096
#define NPC    4194304ll        // per-channel element count = B*H*W

// ---------------- workspace layout (uint32 units) ----------------
#define U_SUM_D    0                    // double[12]  (uints 0..23)
#define U_SUMSQ_D  24                   // double[12]  (uints 24..47)
#define U_MINB     48                   // uint[12] monotone-key min
#define U_MAXB     60                   // uint[12] monotone-key max
#define U_CHIST    72                   // uint[12*1000]
#define U_BELOW    (U_CHIST + 12000)    // uint[12]  (#values with floor(x*10)<0)
#define U_QBS      (U_BELOW + 12)       // uint[24]  refine coarse-bin start per (c,q)
#define U_QBN      (U_QBS + 24)         // uint[24]  refine width in coarse bins
#define U_FHIST    (U_QBN + 24)         // uint[12*2*4096]
#define U_FBELOW   (U_FHIST + 12*2*NFINE) // uint[24]
#define WS_UINTS   (U_FBELOW + 24)      // ~442 KB total

// output layout (floats): mean[12] var[12] std[12] min[12] max[12] count[1]
//                         hist[12000] pct02[12] pct98[12]  => 12085
#define O_MEAN 0
#define O_VAR  12
#define O_STD  24
#define O_MIN  36
#define O_MAX  48
#define O_CNT  60
#define O_HIST 61
#define O_P02  12061
#define O_P98  12073

typedef __attribute__((ext_vector_type(2))) float v2f;
typedef __attribute__((ext_vector_type(8))) float v8f;

// f32 16x16x4 WMMA: D = A*B + C.  B = all-ones => every D column holds row sums.
__device__ __forceinline__ v8f wmma_sum(v2f a, v2f ones, v8f c) {
  return __builtin_amdgcn_wmma_f32_16x16x4_f32(false, a, false, ones,
                                               (short)0, c, false, false);
}

// monotone key: unsigned ordering of key == float ordering of f
__device__ __forceinline__ unsigned fkey(float f) {
  unsigned u = __float_as_uint(f);
  return (u & 0x80000000u) ? ~u : (u | 0x80000000u);
}
__device__ __forceinline__ float funkey(unsigned k) {
  unsigned u = (k & 0x80000000u) ? (k ^ 0x80000000u) : ~k;
  return __uint_as_float(u);
}

// ---------------- kernel 0: init workspace ----------------
__global__ void k_init(unsigned* __restrict__ wsu) {
  int i = blockIdx.x * blockDim.x + threadIdx.x;
  if (i < WS_UINTS) {
    unsigned v = 0u;
    if (i >= U_MINB && i < U_MINB + C_) v = 0xFFFFFFFFu;  // key(+inf-ish) max
    wsu[i] = v;
  }
}

// ---------------- kernel 1: streaming stats + coarse hist ----------------
__global__ void __launch_bounds__(256)
k_pass1(const float* __restrict__ x, unsigned* __restrict__ wsu,
        double* __restrict__ wsd) {
  const int slice = blockIdx.x;
  const int c = slice % C_;
  const float4* __restrict__ xv =
      reinterpret_cast<const float4*>(x + (size_t)slice * HW_);

  __shared__ unsigned lh[BINS_];
  __shared__ float redS[8], redQ[8], redMn[8], redMx[8];
  __shared__ unsigned redB[8];

  for (int i = threadIdx.x; i < BINS_; i += 256) lh[i] = 0u;
  __syncthreads();

  v8f accS = {}; v8f accQ = {};
  v2f ones; ones[0] = 1.0f; ones[1] = 1.0f;
  float mn = INFINITY, mx = -INFINITY;
  unsigned nbelow = 0;

  #pragma unroll 4
  for (int it = 0; it < HW_ / 4 / 256; ++it) {            // 64 iterations
    float4 v = xv[it * 256 + threadIdx.x];
    if (it + 8 < 64) __builtin_prefetch(&xv[(it + 8) * 256 + threadIdx.x], 0, 0);

    v2f a0; a0[0] = v.x;       a0[1] = v.y;
    v2f a1; a1[0] = v.z;       a1[1] = v.w;
    v2f q0; q0[0] = v.x * v.x; q0[1] = v.y * v.y;
    v2f q1; q1[0] = v.z * v.z; q1[1] = v.w * v.w;
    accS = wmma_sum(a0, ones, accS);
    accS = wmma_sum(a1, ones, accS);
    accQ = wmma_sum(q0, ones, accQ);
    accQ = wmma_sum(q1, ones, accQ);

    mn = fminf(mn, fminf(fminf(v.x, v.y), fminf(v.z, v.w)));
    mx = fmaxf(mx, fmaxf(fmaxf(v.x, v.y), fmaxf(v.z, v.w)));

    float vals[4] = {v.x, v.y, v.z, v.w};
    #pragma unroll
    for (int j = 0; j < 4; ++j) {
      float f = vals[j];
      float s = f * 10.0f;
      int iq = (int)floorf(s);
      nbelow += (iq < 0) ? 1u : 0u;
      if (f >= 0.0f && f <= 100.0f) {
        int ib = iq < 0 ? 0 : (iq > BINS_ - 1 ? BINS_ - 1 : iq);
        atomicAdd(&lh[ib], 1u);
      }
    }
  }

  // lane partial from WMMA C/D layout: sum of 8 VGPRs, wave reduce, /16
  float ls = 0.f, lq = 0.f;
  #pragma unroll
  for (int j = 0; j < 8; ++j) { ls += accS[j]; lq += accQ[j]; }
  for (int off = 16; off > 0; off >>= 1) {
    ls += __shfl_xor(ls, off, 32);
    lq += __shfl_xor(lq, off, 32);
    mn = fminf(mn, __shfl_xor(mn, off, 32));
    mx = fmaxf(mx, __shfl_xor(mx, off, 32));
    nbelow += (unsigned)__shfl_xor((int)nbelow, off, 32);
  }
  int wave = threadIdx.x >> 5, lane = threadIdx.x & 31;
  if (lane == 0) {
    redS[wave] = ls * (1.0f / 16.0f);  // each row-sum replicated in 16 lanes
    redQ[wave] = lq * (1.0f / 16.0f);
    redMn[wave] = mn; redMx[wave] = mx; redB[wave] = nbelow;
  }
  __syncthreads();

  if (threadIdx.x == 0) {
    double bs = 0.0, bq = 0.0;
    float bmn = INFINITY, bmx = -INFINITY;
    unsigned bb = 0;
    #pragma unroll
    for (int w = 0; w < 8; ++w) {
      bs += (double)redS[w]; bq += (double)redQ[w];
      bmn = fminf(bmn, redMn[w]); bmx = fmaxf(bmx, redMx[w]);
      bb += redB[w];
    }
    atomicAdd(&wsd[c], bs);
    atomicAdd(&wsd[C_ + c], bq);
    atomicMin(&wsu[U_MINB + c], fkey(bmn));
    atomicMax(&wsu[U_MAXB + c], fkey(bmx));
    if (bb) atomicAdd(&wsu[U_BELOW + c], bb);
  }
  for (int i = threadIdx.x; i < BINS_; i += 256)
    if (lh[i]) atomicAdd(&wsu[U_CHIST + c * BINS_ + i], lh[i]);
}

// ---------------- kernel 2: localize quantile ranks to coarse bins ----------------
__global__ void k_select(unsigned* __restrict__ wsu) {
  int t = threadIdx.x;
  if (t >= 2 * C_) return;
  int c = t >> 1, q = t & 1;
  double qv = q ? 0.98 : 0.02;
  long long k0 = (long long)(qv * (double)(NPC - 1));
  long long k1 = k0 + 1;
  long long cum = (long long)wsu[U_BELOW + c];
  int b0 = (k0 < cum) ? 0 : -1;
  int b1 = (k1 < cum) ? 0 : -1;
  const unsigned* ch = &wsu[U_CHIST + c * BINS_];
  for (int b = 0; b < BINS_ && (b0 < 0 || b1 < 0); ++b) {
    long long cnt = (long long)ch[b];
    if (b0 < 0 && k0 < cum + cnt) b0 = b;
    if (b1 < 0 && k1 < cum + cnt) b1 = b;
    cum += cnt;
  }
  if (b0 < 0) b0 = BINS_ - 1;
  if (b1 < 0) b1 = BINS_ - 1;
  if (b1 < b0) b1 = b0;
  wsu[U_QBS + c * 2 + q] = (unsigned)b0;
  wsu[U_QBN + c * 2 + q] = (unsigned)(b1 - b0 + 1);
}

// ---------------- kernel 3: fine histogram over candidate range ----------------
__global__ void __launch_bounds__(256)
k_pass3(const float* __restrict__ x, unsigned* __restrict__ wsu) {
  const int slice = blockIdx.x;
  const int c = slice % C_;
  const int bs0 = (int)wsu[U_QBS + c * 2 + 0];
  const int nb0 = (int)wsu[U_QBN + c * 2 + 0];
  const int bs1 = (int)wsu[U_QBS + c * 2 + 1];
  const int nb1 = (int)wsu[U_QBN + c * 2 + 1];
  const float sc0 = (float)NFINE / (float)nb0;
  const float sc1 = (float)NFINE / (float)nb1;

  __shared__ unsigned fh[2][NFINE];
  __shared__ unsigned rb0[8], rb1[8];
  for (int i = threadIdx.x; i < 2 * NFINE; i += 256) fh[i >> 12][i & (NFINE - 1)] = 0u;
  __syncthreads();

  const float4* __restrict__ xv =
      reinterpret_cast<const float4*>(x + (size_t)slice * HW_);
  unsigned bl0 = 0, bl1 = 0;

  for (int it = 0; it < HW_ / 4 / 256; ++it) {
    float4 v = xv[it * 256 + threadIdx.x];
    if (it + 8 < 64) __builtin_prefetch(&xv[(it + 8) * 256 + threadIdx.x], 0, 0);
    float vals[4] = {v.x, v.y, v.z, v.w};
    #pragma unroll
    for (int j = 0; j < 4; ++j) {
      float s = vals[j] * 10.0f;
      int iq = (int)floorf(s);
      int r0 = iq - bs0;
      if ((unsigned)r0 < (unsigned)nb0) {
        int fb = (int)((s - (float)bs0) * sc0);
        fb = fb < 0 ? 0 : (fb > NFINE - 1 ? NFINE - 1 : fb);
        atomicAdd(&fh[0][fb], 1u);
      }
      bl0 += (iq < bs0) ? 1u : 0u;
      int r1 = iq - bs1;
      if ((unsigned)r1 < (unsigned)nb1) {
        int fb = (int)((s - (float)bs1) * sc1);
        fb = fb < 0 ? 0 : (fb > NFINE - 1 ? NFINE - 1 : fb);
        atomicAdd(&fh[1][fb], 1u);
      }
      bl1 += (iq < bs1) ? 1u : 0u;
    }
  }

  for (int off = 16; off > 0; off >>= 1) {
    bl0 += (unsigned)__shfl_xor((int)bl0, off, 32);
    bl1 += (unsigned)__shfl_xor((int)bl1, off, 32);
  }
  int wave = threadIdx.x >> 5, lane = threadIdx.x & 31;
  if (lane == 0) { rb0[wave] = bl0; rb1[wave] = bl1; }
  __syncthreads();
  if (threadIdx.x == 0) {
    unsigned t0 = 0, t1 = 0;
    #pragma unroll
    for (int w = 0; w < 8; ++w) { t0 += rb0[w]; t1 += rb1[w]; }
    if (t0) atomicAdd(&wsu[U_FBELOW + c * 2 + 0], t0);
    if (t1) atomicAdd(&wsu[U_FBELOW + c * 2 + 1], t1);
  }
  for (int i = threadIdx.x; i < 2 * NFINE; i += 256) {
    int q = i >> 12, f = i & (NFINE - 1);
    unsigned v = fh[q][f];
    if (v) atomicAdd(&wsu[U_FHIST + (c * 2 + q) * NFINE + f], v);
  }
}

// ---------------- kernel 4: hist output = input + batch ----------------
__global__ void k_hist_out(const float* __restrict__ hin,
                           const unsigned* __restrict__ wsu,
                           float* __restrict__ out) {
  int i = blockIdx.x * blockDim.x + threadIdx.x;
  if (i < C_ * BINS_) out[O_HIST + i] = hin[i] + (float)wsu[U_CHIST + i];
}

// ---------------- kernel 5: finalize scalars + quantiles ----------------
__global__ void k_final(const float* __restrict__ mean_in,
                        const float* __restrict__ var_in,
                        const float* __restrict__ count_in,
                        const float* __restrict__ min_in,
                        const float* __restrict__ max_in,
                        const unsigned* __restrict__ wsu,
                        const double* __restrict__ wsd,
                        float* __restrict__ out) {
  int t = threadIdx.x;
  if (t < C_) {
    int c = t;
    double s = wsd[c], sq = wsd[C_ + c];
    double Nd = (double)NPC;
    double bm = s / Nd;
    double bv = (sq - s * s / Nd) / (Nd - 1.0);   // unbiased
    double cnt = (double)count_in[0];
    double bc = (double)B_;                       // x.shape[0]
    double nab = cnt + bc;
    double m = (double)mean_in[c], va = (double)var_in[c];
    double delta = bm - m;
    double nm = (m * cnt + bm * bc) / nab;
    double nv = (va * cnt + bv * bc + delta * delta * cnt * bc / nab) / nab;
    out[O_MEAN + c] = (float)nm;
    out[O_VAR + c] = (float)nv;
    out[O_STD + c] = (float)sqrt(nv + 1e-8);
    out[O_MIN + c] = fminf(min_in[c], funkey(wsu[U_MINB + c]));
    out[O_MAX + c] = fmaxf(max_in[c], funkey(wsu[U_MAXB + c]));
    if (c == 0) out[O_CNT] = count_in[0] + (float)B_;
  }
  if (t < 2 * C_) {
    int c = t >> 1, q = t & 1;
    double qv = q ? 0.98 : 0.02;
    double k = qv * (double)(NPC - 1);
    long long k0 = (long long)k;
    double frac = k - (double)k0;
    int bs = (int)wsu[U_QBS + c * 2 + q];
    int nb = (int)wsu[U_QBN + c * 2 + q];
    long long cum = (long long)wsu[U_FBELOW + c * 2 + q];
    double lo = (double)bs * 0.1;
    double binw = (double)nb * 0.1 / (double)NFINE;
    double hi = lo + (double)nb * 0.1;
    double v0 = hi, v1 = hi;
    bool h0 = (k0 < cum), h1 = (k0 + 1 < cum);
    if (h0) v0 = lo;
    if (h1) v1 = lo;
    const unsigned* fh = &wsu[U_FHIST + (c * 2 + q) * NFINE];
    for (int f = 0; f < NFINE && !(h0 && h1); ++f) {
      long long cnt2 = (long long)fh[f];
      if (!cnt2) continue;
      if (!h0 && k0 < cum + cnt2) {
        v0 = lo + binw * ((double)f + ((double)(k0 - cum) + 0.5) / (double)cnt2);
        h0 = true;
      }
      if (!h1 && k0 + 1 < cum + cnt2) {
        v1 = lo + binw * ((double)f + ((double)(k0 + 1 - cum) + 0.5) / (double)cnt2);
        h1 = true;
      }
      cum += cnt2;
    }
    out[(q ? O_P98 : O_P02) + c] = (float)(v0 + frac * (v1 - v0));
  }
}

// ---------------- launch ----------------
extern "C" void kernel_launch(void* const* d_in, const int* in_sizes, int n_in,
                              void* d_out, int out_size, void* d_ws, size_t ws_size,
                              hipStream_t stream) {
  const float* x        = (const float*)d_in[0];
  const float* mean_in  = (const float*)d_in[1];
  const float* var_in   = (const float*)d_in[2];
  const float* count_in = (const float*)d_in[3];
  const float* min_in   = (const float*)d_in[4];
  const float* max_in   = (const float*)d_in[5];
  const float* hist_in  = (const float*)d_in[6];
  float*    out = (float*)d_out;
  unsigned* wsu = (unsigned*)d_ws;
  double*   wsd = (double*)d_ws;

  k_init<<<(WS_UINTS + 255) / 256, 256, 0, stream>>>(wsu);
  k_pass1<<<SLICES, 256, 0, stream>>>(x, wsu, wsd);
  k_select<<<1, 32, 0, stream>>>(wsu);
  k_pass3<<<SLICES, 256, 0, stream>>>(x, wsu);
  k_hist_out<<<(C_ * BINS_ + 255) / 256, 256, 0, stream>>>(hist_in, wsu, out);
  k_final<<<1, 32, 0, stream>>>(mean_in, var_in, count_in, min_in, max_in,
                                wsu, wsd, out);
}